// SpeakerLoss_55611236548990
// MI455X (gfx1250) — compile-verified
//
#include <hip/hip_runtime.h>

// ---------------------------------------------------------------------------
// SpeakerLoss on MI455X (gfx1250, wave32):
//   d[i] = ||vs_i - other_vs_i||^2  (D=64)
//   loss = mean( same ? d : max(L - d, 0) )
// Memory-bound streaming reduction (~512MB @ 23.3 TB/s ~ 22us). Row sums are
// computed with chained V_WMMA_F32_16X16X4_F32 (A = diff^2, B = ones), data is
// streamed with non-temporal b128 loads, final reduction is deterministic.
// ---------------------------------------------------------------------------

typedef float f32x4 __attribute__((ext_vector_type(4)));
typedef __attribute__((ext_vector_type(2))) float v2f;
typedef __attribute__((ext_vector_type(8))) float v8f;

static constexpr float L_MARGIN = 1.0f;

__global__ __launch_bounds__(256) void speaker_loss_partial(
    const int*   __restrict__ s,
    const float* __restrict__ vs,
    const int*   __restrict__ other_s,
    const float* __restrict__ other_vs,
    float*       __restrict__ partials,
    int B)
{
    const int lane         = threadIdx.x & 31;
    const int waveInBlock  = threadIdx.x >> 5;
    const int wavesPerBlk  = blockDim.x >> 5;
    const int waveId       = blockIdx.x * wavesPerBlk + waveInBlock;
    const int numWaves     = gridDim.x * wavesPerBlk;

    // A-matrix layout for V_WMMA_F32_16X16X4_F32 (16x4 f32):
    //   lanes 0-15  : M = lane,    VGPR0/1 = K0,K1
    //   lanes 16-31 : M = lane-16, VGPR0/1 = K2,K3
    // B = all-ones 4x16, so any assignment of columns to K slots is valid:
    // lower lanes cover row columns [0,32), upper lanes cover [32,64).
    const int rowInTile = lane & 15;
    const int half      = lane >> 4;   // 0 -> cols 0..31, 1 -> cols 32..63

    const int numTiles = B >> 4;       // 16 rows per tile

    v2f ones; ones.x = 1.0f; ones.y = 1.0f;
    float total = 0.0f;

    for (int tile = waveId; tile < numTiles; tile += numWaves) {
        const size_t r0   = (size_t)(tile << 4) + rowInTile;
        const float* pa   = vs       + r0 * 64 + half * 32;
        const float* pb   = other_vs + r0 * 64 + half * 32;

        // Stream 128B half-row per lane with non-temporal b128 loads,
        // square the differences into registers.
        float dq[32];
        #pragma unroll
        for (int j = 0; j < 8; ++j) {
            f32x4 a = __builtin_nontemporal_load((const f32x4*)(pa + 4 * j));
            f32x4 b = __builtin_nontemporal_load((const f32x4*)(pb + 4 * j));
            #pragma unroll
            for (int e = 0; e < 4; ++e) {
                float df = a[e] - b[e];
                dq[4 * j + e] = df * df;
            }
        }

        // 16 chained WMMAs reduce K=64: every column of D accumulates the
        // row sum. EXEC is full here (uniform control flow per wave).
        v8f c = {};
        #pragma unroll
        for (int i = 0; i < 16; ++i) {
            v2f av; av.x = dq[2 * i]; av.y = dq[2 * i + 1];
            c = __builtin_amdgcn_wmma_f32_16x16x4_f32(
                    /*neg_a=*/false, av, /*neg_b=*/false, ones,
                    /*c_mod=*/(short)0, c, /*reuse_a=*/false, /*reuse_b=*/false);
        }

        // D layout (16x16 f32 C/D): lane 0 holds rows 0..7 in its 8 VGPRs
        // (N=0), lane 16 holds rows 8..15. Only those lanes run the hinge.
        if (rowInTile == 0) {
            const int base = (tile << 4) + (half << 3);
            #pragma unroll
            for (int r = 0; r < 8; ++r) {
                float d     = c[r];
                float hinge = fmaxf(L_MARGIN - d, 0.0f);
                total += (s[base + r] == other_s[base + r]) ? d : hinge;
            }
        }
    }

    // Wave32 reduction (only lanes 0/16 carry data, full xor-tree is cheap).
    #pragma unroll
    for (int off = 16; off > 0; off >>= 1)
        total += __shfl_xor(total, off, 32);

    // Block reduction through LDS -> one partial per block (deterministic).
    __shared__ float red[32];
    if (lane == 0) red[waveInBlock] = total;
    __syncthreads();
    if (threadIdx.x == 0) {
        float bsum = 0.0f;
        for (int w = 0; w < wavesPerBlk; ++w) bsum += red[w];
        partials[blockIdx.x] = bsum;
    }
}

__global__ __launch_bounds__(256) void speaker_loss_finalize(
    const float* __restrict__ partials, int nparts,
    const int*   __restrict__ s,
    const float* __restrict__ vs,
    const int*   __restrict__ other_s,
    const float* __restrict__ other_vs,
    int B,
    float*       __restrict__ out)
{
    const int tid = threadIdx.x;
    float t = 0.0f;
    for (int i = tid; i < nparts; i += blockDim.x)
        t += partials[i];

    // Scalar tail for B % 16 rows (none for B = 1M, kept for generality).
    const int tailStart = (B >> 4) << 4;
    for (int r = tailStart + tid; r < B; r += blockDim.x) {
        float d = 0.0f;
        for (int j = 0; j < 64; ++j) {
            float df = vs[(size_t)r * 64 + j] - other_vs[(size_t)r * 64 + j];
            d += df * df;
        }
        t += (s[r] == other_s[r]) ? d : fmaxf(L_MARGIN - d, 0.0f);
    }

    // Fixed-order tree reduction -> bitwise deterministic across replays.
    __shared__ float red[256];
    red[tid] = t;
    __syncthreads();
    for (int sft = 128; sft > 0; sft >>= 1) {
        if (tid < sft) red[tid] += red[tid + sft];
        __syncthreads();
    }
    if (tid == 0)
        out[0] = red[0] / (float)B;
}

extern "C" void kernel_launch(void* const* d_in, const int* in_sizes, int n_in,
                              void* d_out, int out_size, void* d_ws, size_t ws_size,
                              hipStream_t stream) {
    const int*   s        = (const int*)d_in[0];
    const float* vs       = (const float*)d_in[1];
    const int*   other_s  = (const int*)d_in[2];
    const float* other_vs = (const float*)d_in[3];
    const int    B        = in_sizes[0];

    float* partials = (float*)d_ws;           // NBLOCKS floats of scratch
    const int NBLOCKS  = 1024;                // 8192 wave32s: HBM-saturating
    const int NTHREADS = 256;                 // 8 waves per workgroup

    speaker_loss_partial<<<NBLOCKS, NTHREADS, 0, stream>>>(
        s, vs, other_s, other_vs, partials, B);
    speaker_loss_finalize<<<1, 256, 0, stream>>>(
        partials, NBLOCKS, s, vs, other_s, other_vs, B, (float*)d_out);
}